// GCNII_73933567034044
// MI455X (gfx1250) — compile-verified
//
#include <hip/hip_runtime.h>
#include <math.h>

#define NND 100000
#define NE  1250000
#define FIN 500
#define HH  64
#define CC  40
#define LL  16
#define ALPHA_ 0.1f
#define LAMDA_ 0.5f
#define KC  128   // K-chunk rows of W0 staged in LDS for the input projection

typedef __attribute__((ext_vector_type(2))) float v2f;
typedef __attribute__((ext_vector_type(8))) float v8f;

// ---------------- utility kernels ----------------

__global__ void zero_i32(int* __restrict__ p, int n) {
    int i = blockIdx.x * blockDim.x + threadIdx.x;
    if (i < n) p[i] = 0;
}

__global__ void count_rows(const long long* __restrict__ rows, int* __restrict__ cnt) {
    int e = blockIdx.x * blockDim.x + threadIdx.x;
    if (e < NE) atomicAdd(&cnt[(int)rows[e]], 1);
}

__global__ void compute_dinv(const int* __restrict__ cnt, float* __restrict__ dinv) {
    int i = blockIdx.x * blockDim.x + threadIdx.x;
    if (i < NND) dinv[i] = rsqrtf((float)cnt[i] + 1.0f);   // +1 self loop; deg >= 1 always
}

// exclusive scan of cnt -> row_ofs  (blocks of 1024, Hillis-Steele in LDS)
__global__ __launch_bounds__(256) void scan_blocks(const int* __restrict__ in,
                                                   int* __restrict__ out,
                                                   int* __restrict__ blksum, int n) {
    __shared__ int sh[1024];
    int base = blockIdx.x * 1024;
    for (int i = threadIdx.x; i < 1024; i += 256) {
        int g = base + i;
        sh[i] = (g < n) ? in[g] : 0;
    }
    __syncthreads();
    for (int off = 1; off < 1024; off <<= 1) {
        int v0 = ((int)threadIdx.x + 0 >= off)   ? sh[threadIdx.x + 0   - off] : 0;
        int v1 = ((int)threadIdx.x + 256 >= off) ? sh[threadIdx.x + 256 - off] : 0;
        int v2 = ((int)threadIdx.x + 512 >= off) ? sh[threadIdx.x + 512 - off] : 0;
        int v3 = ((int)threadIdx.x + 768 >= off) ? sh[threadIdx.x + 768 - off] : 0;
        __syncthreads();
        sh[threadIdx.x + 0]   += v0;
        sh[threadIdx.x + 256] += v1;
        sh[threadIdx.x + 512] += v2;
        sh[threadIdx.x + 768] += v3;
        __syncthreads();
    }
    for (int i = threadIdx.x; i < 1024; i += 256) {
        int g = base + i;
        if (g < n) out[g] = (i == 0) ? 0 : sh[i - 1];
    }
    if (threadIdx.x == 0) blksum[blockIdx.x] = sh[1023];
}

__global__ void scan_sums(int* __restrict__ blksum, int nb) {
    __shared__ int sh[128];
    int t = threadIdx.x;
    sh[t] = (t < nb) ? blksum[t] : 0;
    __syncthreads();
    for (int off = 1; off < 128; off <<= 1) {
        int v = (t >= off) ? sh[t - off] : 0;
        __syncthreads();
        sh[t] += v;
        __syncthreads();
    }
    if (t < nb) blksum[t] = (t == 0) ? 0 : sh[t - 1];
}

__global__ void scan_add(int* __restrict__ out, const int* __restrict__ blksum, int n) {
    int base = blockIdx.x * 1024;
    int add = blksum[blockIdx.x];
    for (int i = threadIdx.x; i < 1024; i += 256) {
        int g = base + i;
        if (g < n) out[g] += add;
    }
}

__global__ void scatter_edges(const long long* __restrict__ rows,
                              const long long* __restrict__ cols,
                              const int* __restrict__ row_ofs,
                              int* __restrict__ cursor,
                              const float* __restrict__ dinv,
                              int* __restrict__ col_s,
                              float* __restrict__ norm_s) {
    int e = blockIdx.x * blockDim.x + threadIdx.x;
    if (e >= NE) return;
    int r = (int)rows[e];
    int c = (int)cols[e];
    int p = row_ofs[r] + atomicAdd(&cursor[r], 1);
    col_s[p]  = c;
    norm_s[p] = dinv[r] * dinv[c];
}

// ---------------- sparse propagation ----------------
// blockDim = (64, 4): 64 lanes own one feature each of one row.
__global__ __launch_bounds__(256) void propagate(const float* __restrict__ h,
                                                 const float* __restrict__ x0,
                                                 const int* __restrict__ row_ofs,
                                                 const int* __restrict__ cnt,
                                                 const int* __restrict__ col_s,
                                                 const float* __restrict__ norm_s,
                                                 const float* __restrict__ dinv,
                                                 float* __restrict__ sout) {
    int row = blockIdx.x * 4 + threadIdx.y;
    if (row >= NND) return;
    int f = threadIdx.x;               // 0..63
    float di = dinv[row];
    float acc = di * di * h[row * HH + f];   // self loop
    int j0 = row_ofs[row];
    int j1 = j0 + cnt[row];
    for (int j = j0; j < j1; ++j) {
        int c = col_s[j];              // wave-uniform per j
        acc += norm_s[j] * h[c * HH + f];
    }
    sout[row * HH + f] = (1.0f - ALPHA_) * acc + ALPHA_ * x0[row * HH + f];
}

// ---------------- WMMA GEMM kernels (fp32, 16x16x4) ----------------
// Each wave: one 16-row tile x 64 cols (4 accumulators). Fragment layouts per ISA 7.12.2:
//  A: lane l<16 -> M=l, vgpr0=K+2*hi+0, vgpr1=K+2*hi+1 (hi = lane>>4)
//  B: vgpr0 = row K+2*hi, vgpr1 = row K+2*hi+1, N = n0 + (lane&15)
//  C: reg r -> (M = m0 + r + 8*hi, N = n0 + (lane&15))

__global__ __launch_bounds__(256) void gemm_proj(const float* __restrict__ X,
                                                 const float* __restrict__ W0,
                                                 const float* __restrict__ b0,
                                                 float* __restrict__ out) {
    __shared__ float shW[KC * HH];   // 32 KB: K-chunk of W0
    __shared__ float shb[HH];
    for (int i = threadIdx.x; i < HH; i += 256) shb[i] = b0[i];

    int wave = threadIdx.x >> 5;
    int tile = blockIdx.x * 8 + wave;          // wave-uniform
    bool active = (tile < NND / 16);
    int lane = threadIdx.x & 31, l15 = lane & 15, hi = lane >> 4;
    int m0 = tile * 16;

    v8f acc[4] = {v8f{}, v8f{}, v8f{}, v8f{}};
    const float* arow = active ? (X + (size_t)(m0 + l15) * FIN + 2 * hi) : X;

    for (int k0 = 0; k0 < FIN; k0 += KC) {
        int kc = (FIN - k0 < KC) ? (FIN - k0) : KC;   // 128,128,128,116 (all %4==0)
        __syncthreads();                               // protect prior-chunk reads
        for (int i = threadIdx.x; i < kc * HH; i += 256) shW[i] = W0[k0 * HH + i];
        __syncthreads();
        if (active) {
            for (int k = 0; k < kc; k += 4) {
                v2f a;
                a.x = arow[k0 + k];
                a.y = arow[k0 + k + 1];
                const float* wr0 = shW + (k + 2 * hi) * HH + l15;
                const float* wr1 = wr0 + HH;
#pragma unroll
                for (int nt = 0; nt < 4; ++nt) {
                    v2f b;
                    b.x = wr0[nt * 16];
                    b.y = wr1[nt * 16];
                    acc[nt] = __builtin_amdgcn_wmma_f32_16x16x4_f32(
                        false, a, false, b, (short)0, acc[nt], false, false);
                }
            }
        }
    }
    if (active) {
#pragma unroll
        for (int nt = 0; nt < 4; ++nt) {
            int col = nt * 16 + l15;
            float bb = shb[col];
#pragma unroll
            for (int r = 0; r < 8; ++r) {
                int row = m0 + r + 8 * hi;
                float v = acc[nt][r] + bb;
                out[row * HH + col] = v > 0.0f ? v : 0.0f;
            }
        }
    }
}

__global__ __launch_bounds__(256) void gemm_layer(const float* __restrict__ S,
                                                  const float* __restrict__ W,
                                                  float beta,
                                                  float* __restrict__ out) {
    __shared__ float shW[HH * HH];
    for (int i = threadIdx.x; i < HH * HH; i += 256) shW[i] = W[i];
    __syncthreads();

    int wave = threadIdx.x >> 5;
    int tile = blockIdx.x * 8 + wave;
    if (tile >= NND / 16) return;
    int lane = threadIdx.x & 31, l15 = lane & 15, hi = lane >> 4;
    int m0 = tile * 16;

    v8f acc[4] = {v8f{}, v8f{}, v8f{}, v8f{}};
    const float* arow = S + (size_t)(m0 + l15) * HH + 2 * hi;

#pragma unroll
    for (int k = 0; k < HH; k += 4) {
        v2f a;
        a.x = arow[k];
        a.y = arow[k + 1];
        const float* wr0 = shW + (k + 2 * hi) * HH + l15;
        const float* wr1 = wr0 + HH;
#pragma unroll
        for (int nt = 0; nt < 4; ++nt) {
            v2f b;
            b.x = wr0[nt * 16];
            b.y = wr1[nt * 16];
            acc[nt] = __builtin_amdgcn_wmma_f32_16x16x4_f32(
                false, a, false, b, (short)0, acc[nt], false, false);
        }
    }
    float omb = 1.0f - beta;
#pragma unroll
    for (int nt = 0; nt < 4; ++nt) {
        int col = nt * 16 + l15;
#pragma unroll
        for (int r = 0; r < 8; ++r) {
            int row = m0 + r + 8 * hi;
            float sv = S[row * HH + col];
            float v = beta * acc[nt][r] + omb * sv;
            out[row * HH + col] = v > 0.0f ? v : 0.0f;
        }
    }
}

__global__ __launch_bounds__(256) void gemm_out(const float* __restrict__ Hm,
                                                const float* __restrict__ Wout,
                                                const float* __restrict__ bout,
                                                float* __restrict__ out) {
    __shared__ float shW[HH * 48];   // Wout zero-padded 40 -> 48 cols
    __shared__ float shb[48];
    for (int i = threadIdx.x; i < HH * 48; i += 256) {
        int r = i / 48, c = i % 48;
        shW[i] = (c < CC) ? Wout[r * CC + c] : 0.0f;
    }
    for (int i = threadIdx.x; i < 48; i += 256) shb[i] = (i < CC) ? bout[i] : 0.0f;
    __syncthreads();

    int wave = threadIdx.x >> 5;
    int tile = blockIdx.x * 8 + wave;
    if (tile >= NND / 16) return;
    int lane = threadIdx.x & 31, l15 = lane & 15, hi = lane >> 4;
    int m0 = tile * 16;

    v8f acc[3] = {v8f{}, v8f{}, v8f{}};
    const float* arow = Hm + (size_t)(m0 + l15) * HH + 2 * hi;

#pragma unroll
    for (int k = 0; k < HH; k += 4) {
        v2f a;
        a.x = arow[k];
        a.y = arow[k + 1];
        const float* wr0 = shW + (k + 2 * hi) * 48 + l15;
        const float* wr1 = wr0 + 48;
#pragma unroll
        for (int nt = 0; nt < 3; ++nt) {
            v2f b;
            b.x = wr0[nt * 16];
            b.y = wr1[nt * 16];
            acc[nt] = __builtin_amdgcn_wmma_f32_16x16x4_f32(
                false, a, false, b, (short)0, acc[nt], false, false);
        }
    }
#pragma unroll
    for (int nt = 0; nt < 3; ++nt) {
        int col = nt * 16 + l15;
#pragma unroll
        for (int r = 0; r < 8; ++r) {
            int row = m0 + r + 8 * hi;
            if (col < CC) out[row * CC + col] = acc[nt][r] + shb[col];
        }
    }
}

// ---------------- host launcher ----------------

extern "C" void kernel_launch(void* const* d_in, const int* in_sizes, int n_in,
                              void* d_out, int out_size, void* d_ws, size_t ws_size,
                              hipStream_t stream) {
    (void)in_sizes; (void)n_in; (void)out_size; (void)ws_size;
    const float*     x    = (const float*)d_in[0];
    const long long* ei   = (const long long*)d_in[1];   // int64 per reference
    const float*     W0   = (const float*)d_in[2];
    const float*     b0   = (const float*)d_in[3];
    const float*     Ws   = (const float*)d_in[4];
    const float*     Wout = (const float*)d_in[5];
    const float*     bout = (const float*)d_in[6];
    float*           out  = (float*)d_out;

    char* ws = (char*)d_ws;
    size_t off = 0;
    auto alloc = [&](size_t bytes) -> void* {
        off = (off + 255) & ~(size_t)255;
        void* p = ws + off;
        off += bytes;
        return p;
    };
    int*   cnt     = (int*)alloc((size_t)NND * 4);
    int*   cursor  = (int*)alloc((size_t)NND * 4);
    int*   row_ofs = (int*)alloc((size_t)NND * 4);
    int*   blksum  = (int*)alloc(256 * 4);
    float* dinv    = (float*)alloc((size_t)NND * 4);
    int*   col_s   = (int*)alloc((size_t)NE * 4);
    float* norm_s  = (float*)alloc((size_t)NE * 4);
    float* x0      = (float*)alloc((size_t)NND * HH * 4);
    float* hbuf    = (float*)alloc((size_t)NND * HH * 4);
    float* sbuf    = (float*)alloc((size_t)NND * HH * 4);

    const long long* rows = ei;
    const long long* cols = ei + NE;

    // CSR build
    zero_i32<<<(NND + 255) / 256, 256, 0, stream>>>(cnt, NND);
    zero_i32<<<(NND + 255) / 256, 256, 0, stream>>>(cursor, NND);
    count_rows<<<(NE + 255) / 256, 256, 0, stream>>>(rows, cnt);
    compute_dinv<<<(NND + 255) / 256, 256, 0, stream>>>(cnt, dinv);
    const int nblk = (NND + 1023) / 1024;   // 98
    scan_blocks<<<nblk, 256, 0, stream>>>(cnt, row_ofs, blksum, NND);
    scan_sums<<<1, 128, 0, stream>>>(blksum, nblk);
    scan_add<<<nblk, 256, 0, stream>>>(row_ofs, blksum, NND);
    scatter_edges<<<(NE + 255) / 256, 256, 0, stream>>>(rows, cols, row_ofs, cursor,
                                                        dinv, col_s, norm_s);

    // input projection: x0 = relu(x @ W0 + b0)
    const int gemm_blocks = (NND / 16 + 7) / 8;   // 782
    gemm_proj<<<gemm_blocks, 256, 0, stream>>>(x, W0, b0, x0);

    // 16 GCNII layers
    const float* hin = x0;
    for (int l = 0; l < LL; ++l) {
        propagate<<<(NND + 3) / 4, dim3(64, 4), 0, stream>>>(hin, x0, row_ofs, cnt,
                                                             col_s, norm_s, dinv, sbuf);
        float beta = logf(LAMDA_ / (float)(l + 1) + 1.0f);
        gemm_layer<<<gemm_blocks, 256, 0, stream>>>(sbuf, Ws + (size_t)l * HH * HH,
                                                    beta, hbuf);
        hin = hbuf;
    }

    // output projection
    gemm_out<<<gemm_blocks, 256, 0, stream>>>(hbuf, Wout, bout, out);
}